// gcn_45423574123075
// MI455X (gfx1250) — compile-verified
//
#include <hip/hip_runtime.h>
#include <hip/hip_bf16.h>

// GCN 2-layer forward for MI455X (gfx1250, wave32).
// Layer1 GEMM (200000x128 @ 128x32 f32) runs on V_WMMA_F32_16X16X4_F32.

typedef __attribute__((ext_vector_type(2))) float v2f;
typedef __attribute__((ext_vector_type(8))) float v8f;

#define GN 200000
#define GE 6400000
#define GDIM 128
#define GHID 32
#define GOUT 2
#define NTILES (GN / 16)   // 12500 exact

// ---------------------------------------------------------------- init
__global__ __launch_bounds__(256) void k_init(float* __restrict__ deg,
                                              float* __restrict__ agg1) {
    int i = blockIdx.x * 256 + threadIdx.x;
    if (i < GN * GHID) agg1[i] = 0.0f;
    if (i < GN) deg[i] = 1.0f;            // self-loop
}

// ------------------------------------------------------------- degrees
__global__ __launch_bounds__(256) void k_deg(const int* __restrict__ ei,
                                             float* __restrict__ deg) {
    int e = blockIdx.x * 256 + threadIdx.x;
    if (e >= GE) return;
    int dst = ei[GE + e];
    atomicAdd(&deg[dst], 1.0f);           // integer-valued -> order-exact
}

__global__ __launch_bounds__(256) void k_rsqrt(float* __restrict__ deg) {
    int i = blockIdx.x * 256 + threadIdx.x;
    if (i < GN) deg[i] = rsqrtf(deg[i]);  // deg -> dis in place
}

// --------------------------------------------------- layer1 GEMM (WMMA)
// h1[N,32] = x[N,128] @ W1[128,32].  One wave computes a 16x32 output
// slab as two 16x16 WMMA tiles, K swept in steps of 4 (f32 WMMA).
__global__ __launch_bounds__(256) void k_gemm1(const float* __restrict__ x,
                                               const float* __restrict__ W1,
                                               float* __restrict__ h1) {
    __shared__ float lW[GDIM * GHID];     // 16 KB: whole W1 in LDS
    for (int i = threadIdx.x; i < GDIM * GHID; i += 256) lW[i] = W1[i];
    __syncthreads();

    const int wave = threadIdx.x >> 5;
    const int lane = threadIdx.x & 31;
    const int tile = blockIdx.x * 8 + wave;
    if (tile >= NTILES) return;

    const int m  = lane & 15;             // row within tile / col within B
    const int hi = lane >> 4;             // lane-half select
    const int kb = hi << 1;               // A: K offset 0 or 2

    const float* __restrict__ arow = x + (size_t)(tile * 16 + m) * GDIM;

    v8f acc0 = {};                        // cols 0..15
    v8f acc1 = {};                        // cols 16..31
    for (int k = 0; k < GDIM; k += 4) {
        // A 16x4 frag: VGPR0 = A[m][k+kb], VGPR1 = A[m][k+kb+1]
        v2f a;
        a.x = arow[k + kb];
        a.y = arow[k + kb + 1];
        // B 4x16 frags from LDS: VGPR0 row k+hi, VGPR1 row k+2+hi
        const int r0 = (k + hi) * GHID;
        const int r1 = (k + 2 + hi) * GHID;
        v2f b0, b1;
        b0.x = lW[r0 + m];        b0.y = lW[r1 + m];
        b1.x = lW[r0 + 16 + m];   b1.y = lW[r1 + 16 + m];
        acc0 = __builtin_amdgcn_wmma_f32_16x16x4_f32(
            false, a, false, b0, (short)0, acc0, false, false);
        acc1 = __builtin_amdgcn_wmma_f32_16x16x4_f32(
            false, a, false, b1, (short)0, acc1, false, false);
    }
    // C/D layout: VGPR v -> M = v + 8*hi, N = lane&15
    const int rbase = tile * 16 + (hi << 3);
#pragma unroll
    for (int v = 0; v < 8; ++v) {
        float* o = h1 + (size_t)(rbase + v) * GHID + m;
        o[0]  = acc0[v];
        o[16] = acc1[v];
    }
}

// --------------------------------------------- layer1 edge aggregation
// 8 threads per edge, 4 features each (float4 load, 4 f32 atomics).
__global__ __launch_bounds__(256) void k_agg1(const int* __restrict__ ei,
                                              const float* __restrict__ h1,
                                              const float* __restrict__ dis,
                                              float* __restrict__ agg1) {
    long long t = (long long)blockIdx.x * 256 + threadIdx.x;
    if (t >= (long long)GE * 8) return;
    int e  = (int)(t >> 3);
    int f0 = ((int)t & 7) << 2;
    int src = ei[e];
    int dst = ei[GE + e];
    float w = dis[src] * dis[dst];
    const float4 hv = *(const float4*)(h1 + (size_t)src * GHID + f0);
    float* o = agg1 + (size_t)dst * GHID + f0;
    atomicAdd(o + 0, hv.x * w);
    atomicAdd(o + 1, hv.y * w);
    atomicAdd(o + 2, hv.z * w);
    atomicAdd(o + 3, hv.w * w);
}

// ----------------------- self-loop + bias + ReLU + dropout (in place)
__global__ __launch_bounds__(256) void k_post1(const float* __restrict__ h1,
                                               const float* __restrict__ dis,
                                               const float* __restrict__ b1,
                                               const float* __restrict__ du,
                                               float* __restrict__ agg1) {
    int i = blockIdx.x * 256 + threadIdx.x;
    if (i >= GN * GHID) return;
    int n = i >> 5, f = i & 31;
    float d  = dis[n];
    float v  = agg1[i] + h1[i] * (d * d) + b1[f];
    v = fmaxf(v, 0.0f);                       // ReLU
    v = (du[i] > 0.5f) ? v * 2.0f : 0.0f;     // dropout p=0.5, scale 2
    agg1[i] = v;                              // h1d in place
}

// ------------------------------------------- layer2 GEMM (K=32, OUT=2)
__global__ __launch_bounds__(256) void k_gemm2(const float* __restrict__ h1d,
                                               const float* __restrict__ W2,
                                               float* __restrict__ h2) {
    int n = blockIdx.x * 256 + threadIdx.x;
    if (n >= GN) return;
    const float* r = h1d + (size_t)n * GHID;
    float a0 = 0.0f, a1 = 0.0f;
#pragma unroll
    for (int k = 0; k < GHID; ++k) {
        float v = r[k];
        a0 = fmaf(v, W2[k * GOUT + 0], a0);
        a1 = fmaf(v, W2[k * GOUT + 1], a1);
    }
    h2[(size_t)n * GOUT + 0] = a0;
    h2[(size_t)n * GOUT + 1] = a1;
}

// ---------------------------- layer2 self-loop term + bias into d_out
__global__ __launch_bounds__(256) void k_out_init(const float* __restrict__ h2,
                                                  const float* __restrict__ dis,
                                                  const float* __restrict__ b2,
                                                  float* __restrict__ out) {
    int i = blockIdx.x * 256 + threadIdx.x;
    if (i >= GN * GOUT) return;
    int n = i >> 1, o = i & 1;
    float d = dis[n];
    out[i] = h2[i] * (d * d) + b2[o];
}

// --------------------------------------------- layer2 edge aggregation
__global__ __launch_bounds__(256) void k_agg2(const int* __restrict__ ei,
                                              const float* __restrict__ h2,
                                              const float* __restrict__ dis,
                                              float* __restrict__ out) {
    int e = blockIdx.x * 256 + threadIdx.x;
    if (e >= GE) return;
    int src = ei[e];
    int dst = ei[GE + e];
    float w = dis[src] * dis[dst];
    const float2 hv = *(const float2*)(h2 + (size_t)src * GOUT);
    atomicAdd(&out[(size_t)dst * GOUT + 0], hv.x * w);
    atomicAdd(&out[(size_t)dst * GOUT + 1], hv.y * w);
}

// ---------------------------------------------------------------- host
extern "C" void kernel_launch(void* const* d_in, const int* in_sizes, int n_in,
                              void* d_out, int out_size, void* d_ws, size_t ws_size,
                              hipStream_t stream) {
    const float* x   = (const float*)d_in[0];   // [N,128]
    const int*   ei  = (const int*)  d_in[1];   // [2,E]
    const float* W1  = (const float*)d_in[2];   // [128,32]
    const float* b1  = (const float*)d_in[3];   // [32]
    const float* W2  = (const float*)d_in[4];   // [32,2]
    const float* b2  = (const float*)d_in[5];   // [2]
    const float* du  = (const float*)d_in[6];   // [N,32]
    float* out = (float*)d_out;                 // [N,2]

    // workspace layout (floats): dis[N] | h1[N*32] | agg1[N*32]
    float* dis  = (float*)d_ws;
    float* h1   = dis  + GN;
    float* agg1 = h1   + (size_t)GN * GHID;
    float* h2   = h1;                           // reuse h1 after post1

    const int B = 256;
    k_init    <<<(GN * GHID + B - 1) / B, B, 0, stream>>>(dis, agg1);
    k_deg     <<<(GE + B - 1) / B,        B, 0, stream>>>(ei, dis);
    k_rsqrt   <<<(GN + B - 1) / B,        B, 0, stream>>>(dis);
    k_gemm1   <<<(NTILES + 7) / 8,        B, 0, stream>>>(x, W1, h1);
    {
        long long t = (long long)GE * 8;
        k_agg1<<<(unsigned)((t + B - 1) / B), B, 0, stream>>>(ei, h1, dis, agg1);
    }
    k_post1   <<<(GN * GHID + B - 1) / B, B, 0, stream>>>(h1, dis, b1, du, agg1);
    k_gemm2   <<<(GN + B - 1) / B,        B, 0, stream>>>(agg1, W2, h2);
    k_out_init<<<(GN * GOUT + B - 1) / B, B, 0, stream>>>(h2, dis, b2, out);
    k_agg2    <<<(GE + B - 1) / B,        B, 0, stream>>>(ei, h2, dis, out);
}